// AlphaGammaNet_67482526154793
// MI455X (gfx1250) — compile-verified
//
#include <hip/hip_runtime.h>
#include <hip/hip_bf16.h>
#include <math.h>

typedef float v2f __attribute__((ext_vector_type(2)));
typedef float v8f __attribute__((ext_vector_type(8)));

#define BB 8
#define TT 16
#define NN 128
#define HD 64
#define GK 8     // G+K spline bases
#define FPI 9    // features per input: silu + 8 bases

// ---------------------------------------------------------------- device math

__device__ __forceinline__ float silu_f(float x) {
    return x / (1.0f + __expf(-x));
}

// Uniform grid B-splines, G=5, K=3, grid[i] = (i-3)*0.4 - 1, i in [0,11].
__device__ __forceinline__ void bspl8(float x, float* out) {
    const float h = 0.4f;
    float grid[12];
#pragma unroll
    for (int i = 0; i < 12; ++i) grid[i] = (float)(i - 3) * h - 1.0f;
    float b[11];
#pragma unroll
    for (int i = 0; i < 11; ++i) b[i] = (x >= grid[i] && x < grid[i + 1]) ? 1.0f : 0.0f;
#pragma unroll
    for (int j = 1; j <= 3; ++j) {
        float inv = 1.0f / ((float)j * h);
#pragma unroll
        for (int i = 0; i < 10; ++i) {      // order-j pass needs b[0 .. 10-j]
            if (i < 11 - j)
                b[i] = (x - grid[i]) * inv * b[i] + (grid[i + j + 1] - x) * inv * b[i + 1];
        }
    }
#pragma unroll
    for (int g = 0; g < GK; ++g) out[g] = b[g];
}

// Scalar KAN dot for one output unit o (used where GEMM shapes don't pay off).
__device__ float kan_dot(const float* x, int in_f, int o,
                         const float* bw, const float* sw, const float* sc,
                         int pre_silu) {
    float acc = 0.0f;
    for (int i = 0; i < in_f; ++i) {
        float xi = x[i];
        if (pre_silu) xi = silu_f(xi);
        float bas[GK];
        bspl8(xi, bas);
        const int   oi = o * in_f + i;
        const float* w = sw + oi * GK;
        float sp = 0.0f;
#pragma unroll
        for (int g = 0; g < GK; ++g) sp += bas[g] * w[g];
        acc += silu_f(xi) * bw[oi] + sp * sc[oi];
    }
    return acc;
}

// ---------------------------------------------------------------- WMMA core
// One wave computes a 16x16 fp32 tile with V_WMMA_F32_16X16X4_F32.
// A-frag layout (ISA 7.12.2): lanes 0-15 -> K = k0,k0+1 ; lanes 16-31 -> K = k0+2,k0+3.
// C/D: VGPR v -> row (v + 8*half), col = lane&15.

__device__ __forceinline__ v8f wmma_zero() {
    v8f a = {0.f, 0.f, 0.f, 0.f, 0.f, 0.f, 0.f, 0.f};
    return a;
}

// C = A(16xK, row-major lda) * B^T where Bt is (16 rows x K, row-major ldb)
__device__ __forceinline__ v8f wmma_abt_tile(const float* At, int lda,
                                             const float* Bt, int ldb, int Kd) {
    const int lane = threadIdx.x & 31;
    const int half = lane >> 4;
    const int l    = lane & 15;
    const float* ar = At + l * lda + 2 * half;
    const float* br = Bt + l * ldb + 2 * half;
    v8f acc = wmma_zero();
    for (int k = 0; k < Kd; k += 4) {
        v2f a; a.x = ar[k]; a.y = ar[k + 1];
        v2f b; b.x = br[k]; b.y = br[k + 1];
        acc = __builtin_amdgcn_wmma_f32_16x16x4_f32(false, a, false, b,
                                                    (short)0, acc, false, false);
    }
    return acc;
}

// C = A(16xK, lda) * B(Kx16, ldb); Bt pre-offset to the 16-col tile.
__device__ __forceinline__ v8f wmma_ab_tile(const float* At, int lda,
                                            const float* Bt, int ldb, int Kd) {
    const int lane = threadIdx.x & 31;
    const int half = lane >> 4;
    const int l    = lane & 15;
    const float* ar = At + l * lda + 2 * half;
    v8f acc = wmma_zero();
    for (int k = 0; k < Kd; k += 4) {
        v2f a; a.x = ar[k]; a.y = ar[k + 1];
        v2f b;
        b.x = Bt[(k + 2 * half) * ldb + l];
        b.y = Bt[(k + 1 + 2 * half) * ldb + l];
        acc = __builtin_amdgcn_wmma_f32_16x16x4_f32(false, a, false, b,
                                                    (short)0, acc, false, false);
    }
    return acc;
}

__device__ __forceinline__ void wmma_store(float* Ct, int ldc, v8f acc,
                                           float alpha, const float* bias) {
    const int lane = threadIdx.x & 31;
    const int half = lane >> 4;
    const int l    = lane & 15;
#pragma unroll
    for (int v = 0; v < 8; ++v) {
        float val = alpha * acc[v];
        if (bias) val += bias[l];
        Ct[(v + 8 * half) * ldc + l] = val;
    }
}

// ---------------------------------------------------------------- kernels

// out = KANfeat(X) @ W^T   (generic batched A*B^T, one wave per 16x16 tile)
__global__ void gemm_abt_kernel(const float* A, int lda, int sA,
                                const float* B, int ldb, int sB,
                                float* C, int ldc, int sC,
                                int Kd, float alpha, const float* bias) {
    const int tm = blockIdx.x, tn = blockIdx.y, z = blockIdx.z;
    const float* At = A + (size_t)z * sA + tm * 16 * lda;
    const float* Bt = B + (size_t)z * sB + tn * 16 * ldb;
    __builtin_prefetch(Bt, 0, 3);   // global_prefetch_b8: pull weight tile toward GL2
    v8f acc = wmma_abt_tile(At, lda, Bt, ldb, Kd);
    wmma_store(C + (size_t)z * sC + tm * 16 * ldc + tn * 16, ldc, acc, alpha,
               bias ? bias + tn * 16 : nullptr);
}

// generic batched A*B
__global__ void gemm_ab_kernel(const float* A, int lda, int sA,
                               const float* B, int ldb, int sB,
                               float* C, int ldc, int sC,
                               int Kd, float alpha) {
    const int tm = blockIdx.x, tn = blockIdx.y, z = blockIdx.z;
    const float* At = A + (size_t)z * sA + tm * 16 * lda;
    const float* Bt = B + (size_t)z * sB + tn * 16;
    __builtin_prefetch(Bt, 0, 3);
    v8f acc = wmma_ab_tile(At, lda, Bt, ldb, Kd);
    wmma_store(C + (size_t)z * sC + tm * 16 * ldc + tn * 16, ldc, acc, alpha, nullptr);
}

// attention scores: S[b,hh,q,k] = (Q_head . K_head)/4, per-head strided A*B^T
__global__ void scores_kernel(const float* Q, const float* Kh, float* S) {
    const int tm = blockIdx.x, tn = blockIdx.y, z = blockIdx.z; // z = b*4+hh
    const int b = z >> 2, hh = z & 3;
    const float* At = Q  + (size_t)(b * NN) * HD + hh * 16 + tm * 16 * HD;
    const float* Bt = Kh + (size_t)(b * NN) * HD + hh * 16 + tn * 16 * HD;
    v8f acc = wmma_abt_tile(At, HD, Bt, HD, 16);
    wmma_store(S + (size_t)z * NN * NN + tm * 16 * NN + tn * 16, NN, acc, 0.25f, nullptr);
}

// attn out: O[b,q,hh*16+d] = sum_k S[b,hh,q,k] * V[b,k,hh*16+d]
__global__ void av_kernel(const float* S, const float* V, float* O) {
    const int tm = blockIdx.x, z = blockIdx.z;
    const int b = z >> 2, hh = z & 3;
    const float* At = S + (size_t)z * NN * NN + tm * 16 * NN;
    const float* Bt = V + (size_t)(b * NN) * HD + hh * 16;
    v8f acc = wmma_ab_tile(At, NN, Bt, HD, NN);
    wmma_store(O + (size_t)(b * NN) * HD + hh * 16 + tm * 16 * HD, HD, acc, 1.0f, nullptr);
}

// pack KAN weights -> W[o][i*9+0]=base, W[o][i*9+1+g]=spline*scaler
__global__ void pack_w_kernel(const float* bw, const float* sw, const float* sc,
                              int in_f, int out_f, float* W) {
    int idx = blockIdx.x * blockDim.x + threadIdx.x;
    if (idx >= out_f * in_f) return;
    float* w = W + (size_t)(idx / in_f) * in_f * FPI + (idx % in_f) * FPI;
    w[0] = bw[idx];
    float s = sc[idx];
#pragma unroll
    for (int g = 0; g < GK; ++g) w[1 + g] = sw[idx * GK + g] * s;
}

// feature expansion: feat[row][i*9+..] = {silu(x), B_0..B_7(x)}; x from x1 (i<in1) else x2
__global__ void feat_kernel(const float* x1, const float* x2, int in1, int in_f,
                            float* feat, int rows) {
    int idx = blockIdx.x * blockDim.x + threadIdx.x;
    if (idx >= rows * in_f) return;
    int i = idx % in_f, row = idx / in_f;
    float x = (i < in1) ? x1[(size_t)row * in1 + i]
                        : x2[(size_t)row * (in_f - in1) + (i - in1)];
    float* o = feat + (size_t)row * in_f * FPI + i * FPI;
    o[0] = silu_f(x);
    float bas[GK];
    bspl8(x, bas);
#pragma unroll
    for (int g = 0; g < GK; ++g) o[1 + g] = bas[g];
}

// node encoder: in=4 -> out=64, scalar KAN
__global__ void encode_kernel(const float* xseq, int t, const float* bw,
                              const float* sw, const float* sc, float* ht) {
    int idx = blockIdx.x * blockDim.x + threadIdx.x;
    if (idx >= BB * NN * HD) return;
    int o = idx % HD, row = idx / HD;
    int b = row / NN, n = row % NN;
    const float* x = xseq + (size_t)(((b * TT + t) * NN + n)) * 4;
    ht[(size_t)row * HD + o] = kan_dot(x, 4, o, bw, sw, sc, 0);
}

// GAT energy: kan_force over r_ij = h_i - h_j (out_f=1), masked by adj.
// One block per (b,i) row; h_i and all kan_force params staged in LDS.
__global__ void energy_kernel(const float* ht, const int* adj, int t,
                              const float* bw, const float* sw, const float* sc,
                              float* energy) {
    const int bi = blockIdx.x;          // b*NN + i
    const int j  = threadIdx.x;         // 0..127
    const int b  = bi / NN, i = bi % NN;
    __shared__ float sxi[HD];
    __shared__ float sbw[HD];
    __shared__ float ssc[HD];
    __shared__ float ssw[HD * GK];
    if (j < HD) {
        sxi[j] = ht[(size_t)bi * HD + j];
        sbw[j] = bw[j];
        ssc[j] = sc[j];
    }
    for (int k = j; k < HD * GK; k += NN) ssw[k] = sw[k];
    __syncthreads();

    const float* xj = ht + (size_t)(b * NN + j) * HD;
    float e = 0.0f;
    for (int c = 0; c < HD; ++c) {
        float r = sxi[c] - xj[c];
        float bas[GK];
        bspl8(r, bas);
        float sp = 0.0f;
#pragma unroll
        for (int g = 0; g < GK; ++g) sp += bas[g] * ssw[c * GK + g];
        e += silu_f(r) * sbw[c] + sp * ssc[c];
    }
    int a = adj[(size_t)(((b * TT + t) * NN + i)) * NN + j];
    energy[(size_t)bi * NN + j] = (a == 0) ? -1e9f : e;
}

// row softmax, width 128, in place
__global__ void softmax128_kernel(float* x, int rows) {
    int row = blockIdx.x;
    int tid = threadIdx.x;
    if (row >= rows) return;
    __shared__ float sm[128];
    float v = x[(size_t)row * 128 + tid];
    sm[tid] = v;
    __syncthreads();
    for (int s = 64; s > 0; s >>= 1) {
        if (tid < s) sm[tid] = fmaxf(sm[tid], sm[tid + s]);
        __syncthreads();
    }
    float mx = sm[0];
    __syncthreads();
    float e = __expf(v - mx);
    sm[tid] = e;
    __syncthreads();
    for (int s = 64; s > 0; s >>= 1) {
        if (tid < s) sm[tid] += sm[tid + s];
        __syncthreads();
    }
    x[(size_t)row * 128 + tid] = e / sm[0];
}

__global__ void nodes_mean_kernel(const float* nodes, float* nm) {
    int idx = blockIdx.x * blockDim.x + threadIdx.x;
    if (idx >= BB * HD) return;
    int c = idx % HD, b = idx / HD;
    float s = 0.0f;
    for (int n = 0; n < NN; ++n) s += nodes[(size_t)(b * NN + n) * HD + c];
    nm[idx] = s * (1.0f / NN);
}

// KAN-GRU step (graph level, B=8 rows); h updated in place
__global__ void gru_kernel(const float* nm, float* h,
                           const float* z_bw, const float* z_sw, const float* z_sc,
                           const float* r_bw, const float* r_sw, const float* r_sc,
                           const float* h_bw, const float* h_sw, const float* h_sc) {
    int b = blockIdx.x, o = threadIdx.x;
    __shared__ float comb[2 * HD], combh[2 * HD], hprev[HD];
    comb[o] = nm[b * HD + o];
    float hp = h[b * HD + o];
    comb[HD + o] = hp;
    hprev[o] = hp;
    __syncthreads();
    float zv = 1.0f / (1.0f + __expf(-kan_dot(comb, 2 * HD, o, z_bw, z_sw, z_sc, 0)));
    float rv = 1.0f / (1.0f + __expf(-kan_dot(comb, 2 * HD, o, r_bw, r_sw, r_sc, 0)));
    combh[o] = comb[o];
    combh[HD + o] = rv * hprev[o];
    __syncthreads();
    float htl = tanhf(kan_dot(combh, 2 * HD, o, h_bw, h_sw, h_sc, 0));
    h[b * HD + o] = (1.0f - zv) * hprev[o] + zv * htl;
}

// aq = agent_seq[:, -1] @ agent_w^T + agent_b
__global__ void aq_kernel(const float* agent, const float* w, const float* bv, float* aq) {
    int idx = blockIdx.x * blockDim.x + threadIdx.x;
    if (idx >= BB * NN * HD) return;
    int o = idx % HD, row = idx / HD;
    int b = row / NN, n = row % NN;
    const float* a = agent + (size_t)(((b * TT + (TT - 1)) * NN + n)) * 3;
    aq[idx] = a[0] * w[o * 3 + 0] + a[1] * w[o * 3 + 1] + a[2] * w[o * 3 + 2] + bv[o];
}

// final = mean_n(attnout) + h_temporal
__global__ void final_kernel(const float* attnout, const float* h, float* fin) {
    int idx = blockIdx.x * blockDim.x + threadIdx.x;
    if (idx >= BB * HD) return;
    int c = idx % HD, b = idx / HD;
    float s = 0.0f;
    for (int n = 0; n < NN; ++n) s += attnout[(size_t)(b * NN + n) * HD + c];
    fin[idx] = s * (1.0f / NN) + h[idx];
}

// small scalar KAN (cls heads)
__global__ void kan_small_kernel(const float* x, int in_f, int out_f,
                                 const float* bw, const float* sw, const float* sc,
                                 float* out, int rows, int pre_silu) {
    int idx = blockIdx.x * blockDim.x + threadIdx.x;
    if (idx >= rows * out_f) return;
    int o = idx % out_f, row = idx / out_f;
    out[idx] = kan_dot(x + (size_t)row * in_f, in_f, o, bw, sw, sc, pre_silu);
}

// aux head: silu(final@w1^T+b1)@w2^T+b2
__global__ void aux_kernel(const float* fin, const float* w1, const float* b1,
                           const float* w2, const float* b2, float* out) {
    int row = threadIdx.x;
    if (row >= BB) return;
    float acc = b2[0];
    for (int k = 0; k < 16; ++k) {
        float hsum = b1[k];
        for (int i = 0; i < HD; ++i) hsum += fin[row * HD + i] * w1[k * HD + i];
        acc += silu_f(hsum) * w2[k];
    }
    out[row] = acc;
}

__global__ void copy_kernel(const float* src, float* dst, int n) {
    int i = blockIdx.x * blockDim.x + threadIdx.x;
    if (i < n) dst[i] = src[i];
}

// ---------------------------------------------------------------- launch

extern "C" void kernel_launch(void* const* d_in, const int* in_sizes, int n_in,
                              void* d_out, int out_size, void* d_ws, size_t ws_size,
                              hipStream_t stream) {
    (void)in_sizes; (void)n_in; (void)out_size; (void)ws_size;

    // inputs (recursive insertion-order flatten of setup_inputs())
    const float* x_seq  = (const float*)d_in[0];
    const int*   adj    = (const int*)d_in[1];
    const float* agent  = (const float*)d_in[2];
    // each KAN dict: base_w, spline_w, spline_scaler, grid (grid ignored: analytic)
    const float* enc_bw = (const float*)d_in[3];
    const float* enc_sw = (const float*)d_in[4];
    const float* enc_sc = (const float*)d_in[5];
    const float* f_bw   = (const float*)d_in[7];
    const float* f_sw   = (const float*)d_in[8];
    const float* f_sc   = (const float*)d_in[9];
    const float* m_bw   = (const float*)d_in[11];
    const float* m_sw   = (const float*)d_in[12];
    const float* m_sc   = (const float*)d_in[13];
    const float* u_bw   = (const float*)d_in[15];
    const float* u_sw   = (const float*)d_in[16];
    const float* u_sc   = (const float*)d_in[17];
    const float* z_bw   = (const float*)d_in[19];
    const float* z_sw   = (const float*)d_in[20];
    const float* z_sc   = (const float*)d_in[21];
    const float* r_bw   = (const float*)d_in[23];
    const float* r_sw   = (const float*)d_in[24];
    const float* r_sc   = (const float*)d_in[25];
    const float* h_bw   = (const float*)d_in[27];
    const float* h_sw   = (const float*)d_in[28];
    const float* h_sc   = (const float*)d_in[29];
    const float* agent_w = (const float*)d_in[31];
    const float* agent_b = (const float*)d_in[32];
    const float* in_pw   = (const float*)d_in[33];
    const float* in_pb   = (const float*)d_in[34];
    const float* out_pw  = (const float*)d_in[35];
    const float* out_pb  = (const float*)d_in[36];
    const float* c1_bw   = (const float*)d_in[37];
    const float* c1_sw   = (const float*)d_in[38];
    const float* c1_sc   = (const float*)d_in[39];
    const float* c2_bw   = (const float*)d_in[41];
    const float* c2_sw   = (const float*)d_in[42];
    const float* c2_sc   = (const float*)d_in[43];
    const float* aux_w1  = (const float*)d_in[45];
    const float* aux_b1  = (const float*)d_in[46];
    const float* aux_w2  = (const float*)d_in[47];
    const float* aux_b2  = (const float*)d_in[48];

    const float* wq = in_pw;                const float* bq = in_pb;
    const float* wk = in_pw + 64 * HD;      const float* bk = in_pb + 64;
    const float* wv = in_pw + 128 * HD;     const float* bv = in_pb + 128;

    float* dout = (float*)d_out;

    // workspace carve-out (~12.8 MB fp32)
    float* ws = (float*)d_ws;
    size_t off = 0;
    auto carve = [&](size_t n) { float* p = ws + off; off += n; return p; };
    float* Wmsg  = carve(64 * 576);
    float* Wupd  = carve(64 * 1152);
    float* ht    = carve(BB * NN * HD);
    float* feat1 = carve((size_t)BB * NN * 576);
    float* ener  = carve((size_t)BB * NN * NN);
    float* msg   = carve(BB * NN * HD);
    float* aggr  = carve(BB * NN * HD);
    float* feat2 = carve((size_t)BB * NN * 1152);
    float* nodes = carve(BB * NN * HD);
    float* nmean = carve(BB * HD);
    float* hbuf  = carve(BB * HD);
    float* aq    = carve(BB * NN * HD);
    float* Q     = carve(BB * NN * HD);
    float* Kh    = carve(BB * NN * HD);
    float* V     = carve(BB * NN * HD);
    float* S     = carve((size_t)BB * 4 * NN * NN);
    float* attn  = carve(BB * NN * HD);
    float* attno = carve(BB * NN * HD);
    float* fin   = carve(BB * HD);
    float* y1    = carve(BB * 16);

    hipMemsetAsync(hbuf, 0, BB * HD * sizeof(float), stream);

    // pack KAN weights once per launch
    pack_w_kernel<<<(64 * 64 + 255) / 256, 256, 0, stream>>>(m_bw, m_sw, m_sc, 64, 64, Wmsg);
    pack_w_kernel<<<(64 * 128 + 255) / 256, 256, 0, stream>>>(u_bw, u_sw, u_sc, 128, 64, Wupd);

    for (int t = 0; t < TT; ++t) {
        encode_kernel<<<(BB * NN * HD + 255) / 256, 256, 0, stream>>>(
            x_seq, t, enc_bw, enc_sw, enc_sc, ht);
        energy_kernel<<<BB * NN, NN, 0, stream>>>(ht, adj, t, f_bw, f_sw, f_sc, ener);
        softmax128_kernel<<<BB * NN, 128, 0, stream>>>(ener, BB * NN);
        // msg = KAN(ht): feat(1024x576) @ Wmsg^T
        feat_kernel<<<(BB * NN * 64 + 255) / 256, 256, 0, stream>>>(ht, ht, 64, 64, feat1, BB * NN);
        gemm_abt_kernel<<<dim3(64, 4, 1), 32, 0, stream>>>(
            feat1, 576, 0, Wmsg, 576, 0, msg, 64, 0, 576, 1.0f, nullptr);
        // aggr = alpha @ msg (batched 128x64x128)
        gemm_ab_kernel<<<dim3(8, 4, BB), 32, 0, stream>>>(
            ener, NN, NN * NN, msg, HD, NN * HD, aggr, HD, NN * HD, NN, 1.0f);
        // nodes = KAN([ht, aggr]): feat(1024x1152) @ Wupd^T
        feat_kernel<<<(BB * NN * 128 + 255) / 256, 256, 0, stream>>>(ht, aggr, 64, 128, feat2, BB * NN);
        gemm_abt_kernel<<<dim3(64, 4, 1), 32, 0, stream>>>(
            feat2, 1152, 0, Wupd, 1152, 0, nodes, 64, 0, 1152, 1.0f, nullptr);
        // graph-level KAN-GRU
        nodes_mean_kernel<<<2, 256, 0, stream>>>(nodes, nmean);
        gru_kernel<<<BB, HD, 0, stream>>>(nmean, hbuf,
                                          z_bw, z_sw, z_sc, r_bw, r_sw, r_sc, h_bw, h_sw, h_sc);
    }

    // MHA over last node states
    aq_kernel<<<(BB * NN * HD + 255) / 256, 256, 0, stream>>>(agent, agent_w, agent_b, aq);
    gemm_abt_kernel<<<dim3(64, 4, 1), 32, 0, stream>>>(aq,    HD, 0, wq, HD, 0, Q,  HD, 0, HD, 1.0f, bq);
    gemm_abt_kernel<<<dim3(64, 4, 1), 32, 0, stream>>>(nodes, HD, 0, wk, HD, 0, Kh, HD, 0, HD, 1.0f, bk);
    gemm_abt_kernel<<<dim3(64, 4, 1), 32, 0, stream>>>(nodes, HD, 0, wv, HD, 0, V,  HD, 0, HD, 1.0f, bv);
    scores_kernel<<<dim3(8, 8, BB * 4), 32, 0, stream>>>(Q, Kh, S);
    softmax128_kernel<<<BB * 4 * NN, 128, 0, stream>>>(S, BB * 4 * NN);
    av_kernel<<<dim3(8, 1, BB * 4), 32, 0, stream>>>(S, V, attn);
    gemm_abt_kernel<<<dim3(64, 4, 1), 32, 0, stream>>>(attn, HD, 0, out_pw, HD, 0, attno, HD, 0, HD, 1.0f, out_pb);

    // heads
    final_kernel<<<2, 256, 0, stream>>>(attno, hbuf, fin);
    kan_small_kernel<<<1, 128, 0, stream>>>(fin, 64, 16, c1_bw, c1_sw, c1_sc, y1, BB, 0);
    kan_small_kernel<<<1, 32, 0, stream>>>(y1, 16, 3, c2_bw, c2_sw, c2_sc, dout, BB, 1);
    aux_kernel<<<1, 8, 0, stream>>>(fin, aux_w1, aux_b1, aux_w2, aux_b2, dout + 24);
    copy_kernel<<<2, 256, 0, stream>>>(hbuf, dout + 32, BB * HD);
}